// TextLevelGNN_57277683859507
// MI455X (gfx1250) — compile-verified
//
#include <hip/hip_runtime.h>

// Problem constants (match reference)
#define NUM_NODES 10000
#define DIM       300
#define CLASS_NUM 20
#define BATCH     128
#define SEQ       512
#define KNB       4

typedef __attribute__((ext_vector_type(2))) float v2f;
typedef __attribute__((ext_vector_type(8))) float v8f;

// ---------------------------------------------------------------------------
// ws layout:
//   feats  : float[128][300]  at byte offset 0        (153600 B)
//   logits : float[128][32]   at byte offset 153600   (16384 B, padded N)
// ---------------------------------------------------------------------------

__global__ void zero_feats_kernel(float* __restrict__ feats) {
    int i = blockIdx.x * blockDim.x + threadIdx.x;
    if (i < BATCH * DIM) feats[i] = 0.0f;
}

// Phase 1: gather + edge-weighted K-reduce + node-gated mix + S-reduce.
// Grid (BATCH, 8): each block handles one batch row and 64 sequence positions.
// Thread t owns 2 contiguous f32 of D (coalesced 8B loads of L2-resident rows),
// accumulates over its S-chunk in registers, then one atomicAdd pair.
__global__ __launch_bounds__(160) void gather_reduce_kernel(
    const int*   __restrict__ X,        // [B,S]
    const int*   __restrict__ NX,       // [B,S,K]
    const int*   __restrict__ EW,       // [B,S,K]
    const float* __restrict__ node_emb, // [N,D]
    const float* __restrict__ edge_w,   // [E,1]
    const float* __restrict__ node_w,   // [N,1]
    float*       __restrict__ feats)    // [B,D]
{
    const int b     = blockIdx.x;
    const int chunk = blockIdx.y;          // 8 chunks of 64 positions
    const int tid   = threadIdx.x;         // 0..159, active < 150
    if (tid >= DIM / 2) return;
    const int d2 = tid * 2;

    float accx = 0.0f, accy = 0.0f;
    const int s_begin = chunk * (SEQ / 8);
    const int s_end   = s_begin + (SEQ / 8);

    for (int s = s_begin; s < s_end; ++s) {
        const int   bs = b * SEQ + s;
        const int   x  = X[bs];
        const float nn = node_w[x];        // uniform across block -> broadcast

        float mx = 0.0f, my = 0.0f;
#pragma unroll
        for (int k = 0; k < KNB; ++k) {
            const int   nx = NX[bs * KNB + k];
            const float en = edge_w[EW[bs * KNB + k]];
            const float2 e = *(const float2*)&node_emb[nx * DIM + d2];
            mx += en * e.x;
            my += en * e.y;
        }
        const float2 r = *(const float2*)&node_emb[x * DIM + d2];
        accx += (1.0f - nn) * mx + nn * r.x;
        accy += (1.0f - nn) * my + nn * r.y;
    }
    atomicAdd(&feats[b * DIM + d2],     accx);
    atomicAdd(&feats[b * DIM + d2 + 1], accy);
}

// Phase 2: logits = feats @ fc_w^T + b, ReLU, via V_WMMA_F32_16X16X4_F32.
// One block, 16 waves; wave w -> C tile (m_tile = w>>1, n_tile = w&1).
// A layout (16x4 f32): lanes 0-15 M=0..15 K={0,1}; lanes 16-31 M=0..15 K={2,3}.
// B layout mirrors A (4x16, K rows striped the same way).
// C/D layout: VGPR r holds M = r + 8*(lane>=16), N = lane%16.
__global__ __launch_bounds__(512) void gemm_wmma_kernel(
    const float* __restrict__ feats,   // [128,300]
    const float* __restrict__ fc_w,    // [20,300]
    const float* __restrict__ fc_b,    // [20]
    float*       __restrict__ logits)  // [128,32] padded
{
    const int w    = threadIdx.x >> 5;
    const int lane = threadIdx.x & 31;
    const int mt   = w >> 1;           // 0..7
    const int nt   = w & 1;            // 0..1
    const int lm   = lane & 15;
    const int hi   = lane >> 4;        // 0: K=0,1   1: K=2,3

    const int   row   = mt * 16 + lm;          // A row (feats)
    const int   n     = nt * 16 + lm;          // output column
    const int   nok   = (n < CLASS_NUM);
    const int   ncol  = nok ? n : 0;           // clamp, mask by multiply
    const float bmask = nok ? 1.0f : 0.0f;     // no divergent branch -> EXEC full

    v8f c = {};
#pragma unroll 5
    for (int k0 = 0; k0 < DIM; k0 += 4) {      // 75 K-steps of 4
        const int kk = k0 + 2 * hi;
        v2f a, bm;
        a.x  = feats[row * DIM + kk];
        a.y  = feats[row * DIM + kk + 1];
        bm.x = bmask * fc_w[ncol * DIM + kk];
        bm.y = bmask * fc_w[ncol * DIM + kk + 1];
        c = __builtin_amdgcn_wmma_f32_16x16x4_f32(
                /*neg_a=*/false, a, /*neg_b=*/false, bm,
                /*c_mod=*/(short)0, c, /*reuse_a=*/false, /*reuse_b=*/false);
    }

#pragma unroll
    for (int r = 0; r < 8; ++r) {
        const int m = mt * 16 + hi * 8 + r;
        if (nok) {
            float v = c[r] + fc_b[n];
            logits[m * 32 + n] = v > 0.0f ? v : 0.0f;   // ReLU
        }
    }
}

// Phase 3: row softmax over 20 classes. One thread per batch row.
__global__ void softmax_kernel(const float* __restrict__ logits,
                               float* __restrict__ out)
{
    int b = blockIdx.x * blockDim.x + threadIdx.x;
    if (b >= BATCH) return;
    float v[CLASS_NUM];
    float mx = -1e30f;
#pragma unroll
    for (int n = 0; n < CLASS_NUM; ++n) {
        v[n] = logits[b * 32 + n];
        mx = fmaxf(mx, v[n]);
    }
    float s = 0.0f;
#pragma unroll
    for (int n = 0; n < CLASS_NUM; ++n) {
        v[n] = __expf(v[n] - mx);
        s += v[n];
    }
    const float inv = 1.0f / s;
#pragma unroll
    for (int n = 0; n < CLASS_NUM; ++n)
        out[b * CLASS_NUM + n] = v[n] * inv;
}

extern "C" void kernel_launch(void* const* d_in, const int* in_sizes, int n_in,
                              void* d_out, int out_size, void* d_ws, size_t ws_size,
                              hipStream_t stream) {
    const int*   X        = (const int*)  d_in[0];
    const int*   NX       = (const int*)  d_in[1];
    const int*   EW       = (const int*)  d_in[2];
    const float* node_emb = (const float*)d_in[3];
    const float* edge_w   = (const float*)d_in[4];
    const float* node_w   = (const float*)d_in[5];
    const float* fc_w     = (const float*)d_in[6];
    const float* fc_b     = (const float*)d_in[7];
    float*       out      = (float*)d_out;

    float* feats  = (float*)d_ws;
    float* logits = (float*)((char*)d_ws + (size_t)BATCH * DIM * sizeof(float));

    zero_feats_kernel<<<(BATCH * DIM + 255) / 256, 256, 0, stream>>>(feats);

    gather_reduce_kernel<<<dim3(BATCH, 8), 160, 0, stream>>>(
        X, NX, EW, node_emb, edge_w, node_w, feats);

    gemm_wmma_kernel<<<1, 512, 0, stream>>>(feats, fc_w, fc_b, logits);

    softmax_kernel<<<1, 128, 0, stream>>>(logits, out);
}